// SecondOrderFeatureModule_58059367907751
// MI455X (gfx1250) — compile-verified
//
#include <hip/hip_runtime.h>
#include <math.h>

#define EPS_D 1e-5
#define NTOT (64*64*64)      // 262144 spatial positions
#define C_CH 64
#define ASTR 132             // kernel A LDS row stride (16B-aligned, conflict-free)
#define CSTR 132             // kernel C slab stride
#define PSTR 18              // P scratch stride
#define HSTR 66              // transposed-H stride

typedef __attribute__((ext_vector_type(2))) float v2f;
typedef __attribute__((ext_vector_type(8))) float v8f;
typedef unsigned int u32x4 __attribute__((ext_vector_type(4)));
typedef int i32x4 __attribute__((ext_vector_type(4)));
typedef int i32x8 __attribute__((ext_vector_type(8)));

#if defined(__has_builtin)
#  if __has_builtin(__builtin_amdgcn_tensor_load_to_lds) && \
      __has_builtin(__builtin_amdgcn_tensor_store_from_lds) && \
      __has_builtin(__builtin_amdgcn_s_wait_tensorcnt)
#    define HAVE_TDM 1
#  endif
#endif
#ifndef HAVE_TDM
#  define HAVE_TDM 0
#endif
#if defined(__has_include)
#  if __has_include(<hip/amd_detail/amd_gfx1250_TDM.h>)
#    define TDM_6ARG 1
#  endif
#endif
#ifndef TDM_6ARG
#  define TDM_6ARG 0
#endif

#if HAVE_TDM
// D# for a 64-row x 128-element fp32 tile: iterate mode, one row per
// iteration; LDS rows padded to 132 elements; global rows strided by NTOT.
__device__ __forceinline__ void tdm_build(
    u32x4& g0, i32x8& g1, i32x4& g2, i32x4& g3,
    const void* gaddr, const void* laddr)
{
    unsigned long long ga = (unsigned long long)gaddr;      // global VA
    unsigned la = (unsigned)(unsigned long long)laddr;      // LDS byte offset
    g0.x = 1u;                                              // count=1 (valid)
    g0.y = la;                                              // lds_addr
    g0.z = (unsigned)ga;                                    // global_addr[31:0]
    g0.w = (unsigned)((ga >> 32) & 0x01FFFFFFu) | (2u << 30); // [56:32], type=2
    g1[0] = (2 << 16) | (1 << 19);   // data_size=4B, iterate_enable=1
    g1[1] = (int)(128u << 16);       // tensor_dim0[15:0]=128 (bits 63:48)
    g1[2] = (64 << 16);              // dim0[31:16]=0, tensor_dim1[15:0]=64
    g1[3] = (int)(128u << 16);       // dim1[31:16]=0, tile_dim0=128
    g1[4] = 1;                       // tile_dim1=1, tile_dim2=0
    g1[5] = 128;                     // tensor_dim0_stride[31:0]
    g1[6] = 0;
    g1[7] = 0;
    g2.x = 0;                        // tensor_dim2=0
    g2.y = 132;                      // lds_addr_increment (elements)
    g2.z = NTOT;                     // global_addr_increment (elements)
    g2.w = (63 << 16);               // iterate_count=63 -> 64 rows
    g3.x = 0; g3.y = 0; g3.z = 0; g3.w = 0;
}
__device__ __forceinline__ void tdm_load64x128(float* lds, const float* g) {
    u32x4 d0; i32x8 d1; i32x4 d2, d3;
    tdm_build(d0, d1, d2, d3, (const void*)g, (const void*)lds);
#if TDM_6ARG
    i32x8 d4 = {0, 0, 0, 0, 0, 0, 0, 0};
    __builtin_amdgcn_tensor_load_to_lds(d0, d1, d2, d3, d4, 0);
#else
    __builtin_amdgcn_tensor_load_to_lds(d0, d1, d2, d3, 0);
#endif
}
__device__ __forceinline__ void tdm_store64x128(const float* lds, float* g) {
    u32x4 d0; i32x8 d1; i32x4 d2, d3;
    tdm_build(d0, d1, d2, d3, (const void*)g, (const void*)lds);
#if TDM_6ARG
    i32x8 d4 = {0, 0, 0, 0, 0, 0, 0, 0};
    __builtin_amdgcn_tensor_store_from_lds(d0, d1, d2, d3, d4, 0);
#else
    __builtin_amdgcn_tensor_store_from_lds(d0, d1, d2, d3, 0);
#endif
}
__device__ __forceinline__ void tdm_wait() {
    __builtin_amdgcn_s_wait_tensorcnt(0);
}
#endif // HAVE_TDM

__device__ __forceinline__ v8f wmma4(v2f a, v2f b, v8f c) {
    // D(16x16) = A(16x4) * B(4x16) + C, fp32 throughout
    return __builtin_amdgcn_wmma_f32_16x16x4_f32(
        false, a, false, b, (short)0, c, false, false);
}

// ---------------------------------------------------------------------------
// Kernel 0: zero the gram/sum accumulators in workspace
// ---------------------------------------------------------------------------
__global__ void zero_kernel(float* ws, int n) {
    int i = blockIdx.x * 256 + threadIdx.x;
    if (i < n) ws[i] = 0.0f;
}

// ---------------------------------------------------------------------------
// Kernel 1: raw Gram (sum_n x_c x_d) and channel sums via WMMA f32 16x16x4.
// grid = B * 256 workgroups, 256 threads (8 waves). Each wg: 1024 n in 8
// chunks of 128; slab fill is one TDM descriptor issued by wave 0.
// Each wave owns two adjacent 16x16 Gram tiles (shared A frag).
// ---------------------------------------------------------------------------
__global__ __launch_bounds__(256) void gram_kernel(
    const float* __restrict__ x, float* __restrict__ gram, float* __restrict__ sums)
{
    __shared__ float xs[C_CH * ASTR];
    __shared__ float sred[C_CH];

    const int b    = blockIdx.x >> 8;        // /256
    const int slab = blockIdx.x & 255;
    const int tid  = threadIdx.x;
    const int lane = tid & 31;
    const int wave = tid >> 5;
    const int m15  = lane & 15;
    const int hi   = lane >> 4;              // 0 or 1 (K-half)

    const float* xb = x + (size_t)b * C_CH * NTOT;
    const int n0 = slab * 1024;

    if (tid < C_CH) sred[tid] = 0.0f;

    // tiles: wave owns (ti, tj0) and (ti, tj0+1)
    const int ti  = (2 * wave) >> 2;
    const int tj0 = (2 * wave) & 3;
    v8f acc0 = {}; v8f acc1 = {};

    const int sumC = tid >> 2, sumP = tid & 3;   // 4 threads per channel
    float psum = 0.0f;

    const int rA  = (ti  * 16 + m15) * ASTR + 2 * hi;
    const int rB0 = (tj0 * 16 + m15) * ASTR + 2 * hi;
    const int rB1 = rB0 + 16 * ASTR;

    for (int cs = 0; cs < 1024; cs += 128) {
        __syncthreads();
#if HAVE_TDM
        if (tid < 32) {                      // one DMA for the whole 32KB slab
            tdm_load64x128(xs, &xb[n0 + cs]);
            tdm_wait();
        }
#else
        #pragma unroll
        for (int k = 0; k < 8; ++k) {
            int e = (tid + k * 256) * 4;
            int c = e >> 7;
            int n = e & 127;
            *(float4*)&xs[c * ASTR + n] =
                *(const float4*)&xb[(size_t)c * NTOT + n0 + cs + n];
        }
#endif
        __syncthreads();

        // per-channel partial sums
        #pragma unroll
        for (int j = 0; j < 32; ++j) psum += xs[sumC * ASTR + sumP * 32 + j];

        // Gram accumulate: K runs along n, 4 per WMMA
        #pragma unroll
        for (int kn = 0; kn < 128; kn += 4) {
            v2f a  = *(const v2f*)&xs[rA  + kn];
            v2f b0 = *(const v2f*)&xs[rB0 + kn];
            v2f b1 = *(const v2f*)&xs[rB1 + kn];
            acc0 = wmma4(a, b0, acc0);
            acc1 = wmma4(a, b1, acc1);
        }
    }

    atomicAdd(&sred[sumC], psum);
    __syncthreads();
    if (tid < C_CH) atomicAdd(&sums[b * C_CH + tid], sred[tid]);

    // scatter D fragments: VGPR j -> row (base + j + 8*hi), col = lane&15
    float* gb = gram + (size_t)b * C_CH * C_CH;
    const int rowBase = ti * 16 + hi * 8;
    const int col0    = tj0 * 16 + m15;
    #pragma unroll
    for (int j = 0; j < 8; ++j) {
        atomicAdd(&gb[(rowBase + j) * C_CH + col0],      acc0[j]);
        atomicAdd(&gb[(rowBase + j) * C_CH + col0 + 16], acc1[j]);
    }
}

// ---------------------------------------------------------------------------
// Kernel 2: finalize corr, parallel cyclic-Jacobi eigensolver (fp64 matrix,
// fp32 eigenvectors), top-16 selection, write M = V_topk^T and offsets.
// grid = B, 256 threads.
// ---------------------------------------------------------------------------
__global__ __launch_bounds__(256) void eigh_topk_kernel(
    const float* __restrict__ gram, const float* __restrict__ sums,
    float* __restrict__ Mout, float* __restrict__ offOut)
{
    __shared__ double A[C_CH * C_CH];     // 32 KB
    __shared__ float  Vf[C_CH * C_CH];    // 16 KB
    __shared__ float  mu[C_CH];
    __shared__ float  var[C_CH];
    __shared__ double cs_c[32], cs_s[32];
    __shared__ int    pp[32], qq[32];
    __shared__ double evals[C_CH];
    __shared__ int    topIdx[16];

    const int b   = blockIdx.x;
    const int tid = threadIdx.x;
    const float* G = gram + b * C_CH * C_CH;
    const double Nf = (double)NTOT;

    if (tid < C_CH) mu[tid] = sums[b * C_CH + tid] / (float)NTOT;
    __syncthreads();
    if (tid < C_CH) {
        double m = (double)mu[tid];
        double cov = ((double)G[tid * C_CH + tid] - Nf * m * m) / (Nf - 1.0);
        double d = cov < EPS_D ? EPS_D : cov;
        var[tid] = (float)sqrt(d);
    }
    __syncthreads();
    for (int i = tid; i < C_CH * C_CH; i += 256) {
        int r = i >> 6, c = i & 63;
        double cov  = ((double)G[i] - Nf * (double)mu[r] * (double)mu[c]) / (Nf - 1.0);
        double corr = cov / ((double)var[r] * (double)var[c]);
        if (corr != corr) corr = 0.0;     // nan_to_num
        A[i]  = corr;
        Vf[i] = (r == c) ? 1.0f : 0.0f;
    }
    __syncthreads();

    // 8 sweeps of 63 tournament rounds, 32 disjoint rotations per round
    for (int sweep = 0; sweep < 8; ++sweep) {
        for (int round = 0; round < 63; ++round) {
            if (tid < 32) {
                int i = tid;
                int p = (i == 0) ? 0 : (1 + ((i - 1 + round) % 63));
                int q = 1 + ((62 - i + round) % 63);
                if (p > q) { int t = p; p = q; q = t; }
                pp[tid] = p; qq[tid] = q;
                double apq = A[p * 64 + q];
                double c = 1.0, s = 0.0;
                if (fabs(apq) > 1e-300) {
                    double app = A[p * 64 + p], aqq = A[q * 64 + q];
                    double th = (aqq - app) / (2.0 * apq);
                    double t  = 1.0 / (fabs(th) + sqrt(1.0 + th * th));
                    if (th < 0.0) t = -t;
                    c = 1.0 / sqrt(1.0 + t * t);
                    s = t * c;
                }
                cs_c[tid] = c; cs_s[tid] = s;
            }
            __syncthreads();
            {   // row update: A' = J^T A   (pair i = tid>>3, 8 cols each)
                int i = tid >> 3, j0 = (tid & 7) * 8;
                int p = pp[i], q = qq[i];
                double c = cs_c[i], s = cs_s[i];
                for (int j = j0; j < j0 + 8; ++j) {
                    double ap = A[p * 64 + j], aq = A[q * 64 + j];
                    A[p * 64 + j] = c * ap - s * aq;
                    A[q * 64 + j] = s * ap + c * aq;
                }
            }
            __syncthreads();
            {   // col update: A'' = A' J, and V = V J (fp32)
                int i = tid >> 3, j0 = (tid & 7) * 8;
                int p = pp[i], q = qq[i];
                double c = cs_c[i], s = cs_s[i];
                float cf = (float)c, sf = (float)s;
                for (int j = j0; j < j0 + 8; ++j) {
                    double ap = A[j * 64 + p], aq = A[j * 64 + q];
                    A[j * 64 + p] = c * ap - s * aq;
                    A[j * 64 + q] = s * ap + c * aq;
                    float vp = Vf[j * 64 + p], vq = Vf[j * 64 + q];
                    Vf[j * 64 + p] = cf * vp - sf * vq;
                    Vf[j * 64 + q] = sf * vp + cf * vq;
                }
            }
            __syncthreads();
        }
    }

    if (tid < C_CH) evals[tid] = A[tid * 64 + tid];
    __syncthreads();
    if (tid == 0) {   // top-16 descending (matches top_k order)
        unsigned long long used = 0;
        for (int k = 0; k < 16; ++k) {
            int best = 0; double bv = -1e300;
            for (int c = 0; c < 64; ++c)
                if (!((used >> c) & 1) && evals[c] > bv) { bv = evals[c]; best = c; }
            used |= 1ull << best;
            topIdx[k] = best;
        }
    }
    __syncthreads();
    // M[k][c] = V[c][idx_k];  off[k] = -sum_c M[k][c]*mu[c]
    for (int i = tid; i < 16 * 64; i += 256) {
        int k = i >> 6, c = i & 63;
        Mout[b * 1024 + i] = Vf[c * 64 + topIdx[k]];
    }
    if (tid < 16) {
        float o = 0.0f;
        int idx = topIdx[tid];
        for (int c = 0; c < 64; ++c) o -= Vf[c * 64 + idx] * mu[c];
        offOut[b * 16 + tid] = o;
    }
}

// ---------------------------------------------------------------------------
// Kernel 3: fused copy + projection + MLP, all WMMA. grid = B*256, 256 thr.
// Per iter: TDM load slab 64x128 -> LDS; passthrough copy to out is a TDM
// store overlapping the P-phase WMMAs; each wave does a 16-col tile:
// P(16x16)=M*X+off, H(64x16)=relu(W1*P+b1), ACT(16x16)=W2*H+b2.
// ---------------------------------------------------------------------------
__global__ __launch_bounds__(256) void fused_proj_mlp_kernel(
    const float* __restrict__ x, const float* __restrict__ Mmat,
    const float* __restrict__ offv, const float* __restrict__ w1,
    const float* __restrict__ b1, const float* __restrict__ w2,
    const float* __restrict__ b2, float* __restrict__ out)
{
    __shared__ float slab[C_CH * CSTR];          // 33792 B; reused for H^T
    __shared__ float Pls[8 * 16 * PSTR];         // 9216 B

    const int b   = blockIdx.x >> 8;
    const int seg = blockIdx.x & 255;
    const int tid = threadIdx.x;
    const int lane = tid & 31;
    const int wave = tid >> 5;
    const int m15 = lane & 15;
    const int hi  = lane >> 4;

    const float* xb = x + (size_t)b * C_CH * NTOT;
    float* ob = out + (size_t)b * 80 * NTOT;

    // ---- preload weight fragments into registers ----
    v2f aP[16];                               // M (16x64): A-operand of P
    const float* Mb = Mmat + b * 1024;
    #pragma unroll
    for (int k = 0; k < 16; ++k)
        aP[k] = *(const v2f*)&Mb[m15 * 64 + 4 * k + 2 * hi];
    v8f pInit;
    #pragma unroll
    for (int j = 0; j < 8; ++j) pInit[j] = offv[b * 16 + j + 8 * hi];

    v2f aH[4][4];                             // W1 (64x16): A-operand of H
    #pragma unroll
    for (int t = 0; t < 4; ++t)
        #pragma unroll
        for (int k = 0; k < 4; ++k)
            aH[t][k] = *(const v2f*)&w1[(t * 16 + m15) * 16 + 4 * k + 2 * hi];
    v8f hInit[4];
    #pragma unroll
    for (int t = 0; t < 4; ++t)
        #pragma unroll
        for (int j = 0; j < 8; ++j) hInit[t][j] = b1[t * 16 + j + 8 * hi];

    v2f aA[16];                               // W2 (16x64): A-operand of ACT
    #pragma unroll
    for (int k = 0; k < 16; ++k)
        aA[k] = *(const v2f*)&w2[m15 * 64 + 4 * k + 2 * hi];
    v8f aInit;
    #pragma unroll
    for (int j = 0; j < 8; ++j) aInit[j] = b2[j + 8 * hi];

    float* Pw = &Pls[wave * 16 * PSTR];
    float* Hw = &slab[wave * 16 * HSTR];      // H^T overlay: 8*1056 = 8448 ok
    const int nb = wave * 16;

    for (int it = 0; it < 8; ++it) {
        const int n0 = seg * 1024 + it * 128;
        __syncthreads();
#if HAVE_TDM
        if (tid < 32) {
            tdm_load64x128(slab, &xb[n0]);
            tdm_wait();
        }
#else
        #pragma unroll
        for (int k = 0; k < 8; ++k) {
            int e = (tid + k * 256) * 4;
            int c = e >> 7;
            int n = e & 127;
            *(float4*)&slab[c * CSTR + n] =
                *(const float4*)&xb[(size_t)c * NTOT + n0 + n];
        }
#endif
        __syncthreads();

        // passthrough copy x -> out channels [0,64): TDM store LDS->global,
        // overlaps with the P-phase WMMAs below
#if HAVE_TDM
        if (tid < 32) tdm_store64x128(slab, &ob[n0]);
#else
        #pragma unroll
        for (int k = 0; k < 8; ++k) {
            int e = (tid + k * 256) * 4;
            int c = e >> 7;
            int n = e & 127;
            *(float4*)&ob[(size_t)c * NTOT + n0 + n] =
                *(const float4*)&slab[c * CSTR + n];
        }
#endif

        // P = M * X + off   (K = channel, 16 steps of 4)
        v8f p = pInit;
        #pragma unroll
        for (int kk = 0; kk < 64; kk += 4) {
            v2f bf;
            bf.x = slab[(kk + 2 * hi)     * CSTR + nb + m15];
            bf.y = slab[(kk + 2 * hi + 1) * CSTR + nb + m15];
            p = wmma4(aP[kk >> 2], bf, p);
        }
        #pragma unroll
        for (int j = 0; j < 8; ++j) Pw[(j + 8 * hi) * PSTR + m15] = p[j];
#if HAVE_TDM
        if (tid < 32) tdm_wait();  // out-copy DMA must finish reading slab
#endif
        __syncthreads();           // all waves done with slab; H^T may overwrite

        // H = relu(W1 * P + b1), stored transposed for b64 B-frag loads
        #pragma unroll
        for (int t = 0; t < 4; ++t) {
            v8f h = hInit[t];
            #pragma unroll
            for (int kk = 0; kk < 16; kk += 4) {
                v2f bf;
                bf.x = Pw[(kk + 2 * hi)     * PSTR + m15];
                bf.y = Pw[(kk + 2 * hi + 1) * PSTR + m15];
                h = wmma4(aH[t][kk >> 2], bf, h);
            }
            #pragma unroll
            for (int j = 0; j < 8; ++j) {
                float v = h[j] > 0.0f ? h[j] : 0.0f;
                Hw[m15 * HSTR + t * 16 + j + 8 * hi] = v;   // H^T[n][hid]
            }
        }

        // ACT = W2 * H + b2   (K = hidden, 16 steps of 4; B-frags are b64)
        v8f a = aInit;
        #pragma unroll
        for (int kk = 0; kk < 64; kk += 4) {
            v2f bf = *(const v2f*)&Hw[m15 * HSTR + kk + 2 * hi];
            a = wmma4(aA[kk >> 2], bf, a);
        }
        #pragma unroll
        for (int j = 0; j < 8; ++j)
            ob[(size_t)(64 + j + 8 * hi) * NTOT + n0 + nb + m15] = a[j];
    }
}

// ---------------------------------------------------------------------------
extern "C" void kernel_launch(void* const* d_in, const int* in_sizes, int n_in,
                              void* d_out, int out_size, void* d_ws, size_t ws_size,
                              hipStream_t stream) {
    const float* x  = (const float*)d_in[0];
    const float* w1 = (const float*)d_in[1];
    const float* b1 = (const float*)d_in[2];
    const float* w2 = (const float*)d_in[3];
    const float* b2 = (const float*)d_in[4];
    float* out = (float*)d_out;
    float* ws  = (float*)d_ws;

    float* gram = ws;            // 2 * 64*64
    float* sums = ws + 8192;     // 2 * 64
    float* Mmat = ws + 8320;     // 2 * 16*64
    float* offv = ws + 10368;    // 2 * 16

    zero_kernel<<<(8320 + 255) / 256, 256, 0, stream>>>(ws, 8320);
    gram_kernel<<<2 * 256, 256, 0, stream>>>(x, gram, sums);
    eigh_topk_kernel<<<2, 256, 0, stream>>>(gram, sums, Mmat, offv);
    fused_proj_mlp_kernel<<<2 * 256, 256, 0, stream>>>(x, Mmat, offv, w1, b1, w2, b2, out);
}